// SOG_82549271429519
// MI455X (gfx1250) — compile-verified
//
#include <hip/hip_runtime.h>

// ---------------------------------------------------------------------------
// Fused 3D Sobel-of-Sobel edge loss for MI455X (gfx1250, wave32).
//  - lane == channel (32 fp32 channels contiguous -> 128B coalesced lines)
//  - 4^3 output tile per 256-thread block, 8^3 input halo staged in LDS via
//    gfx1250 async global->LDS DMA (ASYNCcnt), hr tile prefetched behind the
//    pred compute (double buffered: 2*64KB in + 27KB mag = ~155KB LDS/WG,
//    2 workgroups per 320KB WGP).
//  - row-based separable Sobel: one x-row of outputs per inner block, each
//    halo row loaded once (ds_load pairs), zero-coef taps dropped at compile
//    time via integer guards (no reliance on fast-math 0.0*x folding).
//  - deterministic reductions; final 6750-partial sum folded on the matrix
//    pipe via V_WMMA_F32_16X16X4_F32 (sum == 1^T A 1, 64 adds/instruction).
// ---------------------------------------------------------------------------

#define TOUT  4            // output tile edge
#define THALO 8            // input halo edge  (TOUT + 4)
#define TMAG  6            // mag tile edge    (TOUT + 2)
#define CH    32           // channels (== lanes)
#define NT    15           // 60 / TOUT tiles per spatial dim
#define STRD  (64*64*32)   // d stride in floats
#define STRH  (64*32)      // h stride in floats
#define STRW  32           // w stride in floats

#ifdef __HIP_DEVICE_COMPILE__
#  if __has_builtin(__builtin_amdgcn_global_load_async_to_lds_b128)
#    define USE_ASYNC 1
#  else
#    define USE_ASYNC 0
#  endif
#  if __has_builtin(__builtin_amdgcn_s_wait_asynccnt)
#    define HAVE_WAIT_BUILTIN 1
#  else
#    define HAVE_WAIT_BUILTIN 0
#  endif
#else
#  define USE_ASYNC 0
#  define HAVE_WAIT_BUILTIN 0
#endif

typedef int   v4i __attribute__((ext_vector_type(4)));
typedef float v4f __attribute__((ext_vector_type(4)));
typedef float v2f __attribute__((ext_vector_type(2)));
typedef float v8f __attribute__((ext_vector_type(8)));

__device__ __forceinline__ void lds_copy16(const float* g, float* l)
{
#if USE_ASYNC
  // gfx1250: GLOBAL_LOAD_ASYNC_TO_LDS_B128 (tracked with ASYNCcnt, no VGPR data)
  __builtin_amdgcn_global_load_async_to_lds_b128(
      (__attribute__((address_space(1))) v4i*)(g),
      (__attribute__((address_space(3))) v4i*)(l),
      /*offset=*/0, /*cpol=*/0);
#else
  *(v4f*)l = *(const v4f*)g;
#endif
}

__device__ __forceinline__ void wait_async_le16()
{
#if USE_ASYNC
#  if HAVE_WAIT_BUILTIN
  __builtin_amdgcn_s_wait_asynccnt(16);
#  else
  asm volatile("s_wait_asynccnt 0x10" ::: "memory");
#  endif
#endif
}

__device__ __forceinline__ void wait_async_le0()
{
#if USE_ASYNC
#  if HAVE_WAIT_BUILTIN
  __builtin_amdgcn_s_wait_asynccnt(0);
#  else
  asm volatile("s_wait_asynccnt 0x0" ::: "memory");
#  endif
#endif
}

// One x-row of Sobel outputs (NOUT points) from an N^3 LDS cube at row (y,z).
// Derivative-axis sign is irrelevant downstream (squares/abs), so
// conv-vs-crosscorrelation flip does not matter.
// Coefficients: D = {-1,0,1}, S = {1,2,1}. Zero-coef taps are removed by the
// integer guards (compile-time after full unrolling); remaining coefs are
// +/-1 and +/-2, which fold to add/sub/fma without fast-math.
template <int N, int NOUT>
__device__ __forceinline__ void sobel_row(const float* __restrict__ s,
                                          int y, int z, int lane,
                                          float gx[NOUT], float gy[NOUT],
                                          float gz[NOUT])
{
  const float S[3] = {1.f, 2.f, 1.f};
  const float D[3] = {-1.f, 0.f, 1.f};
#pragma unroll
  for (int x = 0; x < NOUT; ++x) { gx[x] = 0.f; gy[x] = 0.f; gz[x] = 0.f; }
#pragma unroll
  for (int a = 0; a < 3; ++a) {
#pragma unroll
    for (int b = 0; b < 3; ++b) {
      const float* p = s + (((z + a) * N + (y + b)) * N) * CH + lane;
      float v[NOUT + 2];
#pragma unroll
      for (int x = 0; x < NOUT + 2; ++x) v[x] = p[x * CH];   // one halo row
#pragma unroll
      for (int x = 0; x < NOUT; ++x) {
        float rs = v[x] + 2.f * v[x + 1] + v[x + 2];  // smooth along w
        float rd = v[x + 2] - v[x];                   // derivative along w
        if (a != 1) gx[x] += (D[a] * S[b]) * rs;      // D[1] == 0
        if (b != 1) gy[x] += (S[a] * D[b]) * rs;      // D[1] == 0
        gz[x] += (S[a] * S[b]) * rd;
      }
    }
  }
}

// One tensor phase: halo cube (THALO^3) -> mag cube (TMAG^3, in LDS) ->
// 8 g2 outputs per wave in registers.
__device__ __forceinline__ void phase_compute(const float* __restrict__ sIn,
                                              float* __restrict__ sMag,
                                              int lane, int wave,
                                              float out[8])
{
  // mag: 36 x-rows of 6 points, 4-5 rows per wave
  for (int r = wave; r < TMAG * TMAG; r += 8) {
    int mz = r / TMAG, my = r % TMAG;
    float gx[TMAG], gy[TMAG], gz[TMAG];
    sobel_row<THALO, TMAG>(sIn, my, mz, lane, gx, gy, gz);
#pragma unroll
    for (int x = 0; x < TMAG; ++x)
      sMag[((mz * TMAG + my) * TMAG + x) * CH + lane] =
          sqrtf(gx[x] * gx[x] + gy[x] * gy[x] + gz[x] * gz[x]);
  }
  __syncthreads();
  // g2: 16 x-rows of 4 points, 2 rows per wave
#pragma unroll
  for (int i = 0; i < 2; ++i) {
    int r = wave * 2 + i;
    int oz = r >> 2, oy = r & 3;
    float gx[TOUT], gy[TOUT], gz[TOUT];
    sobel_row<TMAG, TOUT>(sMag, oy, oz, lane, gx, gy, gz);
#pragma unroll
    for (int x = 0; x < TOUT; ++x)
      out[i * TOUT + x] =
          sqrtf(gx[x] * gx[x] + gy[x] * gy[x]) + gz[x] * gz[x];  // reference
  }
}

__global__ __launch_bounds__(256)
void sobel_g2_loss_kernel(const float* __restrict__ pred,
                          const float* __restrict__ hr,
                          float* __restrict__ part)
{
  extern __shared__ char smem_raw[];
  float* sIn0 = (float*)smem_raw;                         // 16384 floats
  float* sIn1 = sIn0 + THALO * THALO * THALO * CH;        // 16384 floats
  float* sMag = sIn1 + THALO * THALO * THALO * CH;        //  6912 floats
  float* sRed = sMag + TMAG * TMAG * TMAG * CH;           //     8 floats

  const int lane = threadIdx.x & 31;                      // channel
  const int wave = threadIdx.x >> 5;                      // 8 waves

  // grid = (NT, NT, 2*NT): x->tw, y->th, z->b*NT+td (no integer divisions)
  const int tw = blockIdx.x;
  const int th = blockIdx.y;
  const int b  = (blockIdx.z >= NT) ? 1 : 0;
  const int td = blockIdx.z - b * NT;

  const int base = b * (64 * 64 * 64 * 32)
                 + (4 * td) * STRD + (4 * th) * STRH + (4 * tw) * STRW;
  const float* g0 = pred + base;
  const float* g1 = hr + base;

  // Stage both 8x8x8x32ch halo tiles. 128 contiguous 512B chunks per tile,
  // 16 per wave per tensor: 16B per lane per async op (in-order per wave).
  for (int c = wave; c < 128; c += 8) {
    int row = c >> 1, half = c & 1;
    int dz = row >> 3, dy = row & 7;
    int gof = dz * STRD + dy * STRH + half * 128 + lane * 4;
    int lof = ((dz * 8 + dy) * 8) * CH + half * 128 + lane * 4;
    lds_copy16(g0 + gof, sIn0 + lof);
  }
  for (int c = wave; c < 128; c += 8) {                   // hr prefetch
    int row = c >> 1, half = c & 1;
    int dz = row >> 3, dy = row & 7;
    int gof = dz * STRD + dy * STRH + half * 128 + lane * 4;
    int lof = ((dz * 8 + dy) * 8) * CH + half * 128 + lane * 4;
    lds_copy16(g1 + gof, sIn1 + lof);
  }

  wait_async_le16();          // first 16 (pred tile) of 32 ops complete
  __syncthreads();

  float outP[8];
  phase_compute(sIn0, sMag, lane, wave, outP);

  wait_async_le0();           // hr tile complete (overlapped with pred math)
  __syncthreads();            // also fences sMag reuse

  float outH[8];
  phase_compute(sIn1, sMag, lane, wave, outH);

  float acc = 0.f;
#pragma unroll
  for (int i = 0; i < 8; ++i) acc += fabsf(outP[i] - outH[i]);

  // wave32 reduction + cross-wave via LDS (deterministic)
#pragma unroll
  for (int off = 16; off > 0; off >>= 1)
    acc += __shfl_down(acc, off, 32);
  if (lane == 0) sRed[wave] = acc;
  __syncthreads();
  if (threadIdx.x == 0) {
    float s = 0.f;
#pragma unroll
    for (int w = 0; w < 8; ++w) s += sRed[w];
    part[((int)blockIdx.z * NT + (int)blockIdx.y) * NT + (int)blockIdx.x] = s;
  }
}

// Single-wave finalize: total = 1^T * A * 1 on the matrix pipe.
// Pack 64 partials/iteration as the 16x4 f32 A-matrix (2 VGPRs; the lane ->
// (M,K) assignment is irrelevant for a total sum), B = all-ones 4x16
// (layout-independent), accumulate C with V_WMMA_F32_16X16X4_F32.
// Sum over all 16 C columns = 16 * total, so divide once at the end.
// Fixed accumulation order -> bit-deterministic. EXEC all ones (32 threads).
__global__ __launch_bounds__(32)
void finalize_kernel(const float* __restrict__ part, float* __restrict__ out,
                     int n, float scale)
{
  const int lane = threadIdx.x;
  v8f c = {};                         // C accumulator = 0
  v2f bones;
  bones.x = 1.f;
  bones.y = 1.f;

  const int iters = (n + 63) / 64;    // 106 for n = 6750
  for (int it = 0; it < iters; ++it) {
    int i0 = it * 64 + lane * 2;
    v2f a;
    a.x = (i0     < n) ? part[i0]     : 0.f;   // zero-pad tail (sum-safe)
    a.y = (i0 + 1 < n) ? part[i0 + 1] : 0.f;
    c = __builtin_amdgcn_wmma_f32_16x16x4_f32(
        /*neg_a=*/false, a, /*neg_b=*/false, bones,
        /*c_mod=*/(short)0, c, /*reuse_a=*/false, /*reuse_b=*/false);
  }

  float s = 0.f;
#pragma unroll
  for (int r = 0; r < 8; ++r) s += c[r];       // this lane's C elements
#pragma unroll
  for (int off = 16; off > 0; off >>= 1)       // sum across all 32 lanes
    s += __shfl_down(s, off, 32);
  if (lane == 0) out[0] = s * (scale / 16.f);  // 16 replicated columns
}

extern "C" void kernel_launch(void* const* d_in, const int* in_sizes, int n_in,
                              void* d_out, int out_size, void* d_ws, size_t ws_size,
                              hipStream_t stream)
{
  (void)in_sizes; (void)n_in; (void)out_size; (void)ws_size;
  const float* pred = (const float*)d_in[0];
  const float* hr   = (const float*)d_in[1];
  float* part = (float*)d_ws;                 // 6750 partial sums (27 KB)

  const dim3 grid(NT, NT, 2 * NT);            // 15 x 15 x 30 = 6750 blocks
  const int nblocks = 2 * NT * NT * NT;
  const size_t shmem =
      (size_t)(2 * THALO * THALO * THALO * CH + TMAG * TMAG * TMAG * CH + 8) *
      sizeof(float);                          // 158,752 B (<320KB WGP LDS)

  sobel_g2_loss_kernel<<<grid, 256, shmem, stream>>>(pred, hr, part);
  finalize_kernel<<<1, 32, 0, stream>>>(part, (float*)d_out, nblocks,
                                        1.0f / 13824000.0f);
}